// SimpleDirectedCTSGCNLayer_24180665876677
// MI455X (gfx1250) — compile-verified
//
#include <hip/hip_runtime.h>
#include <cstdint>

typedef __attribute__((ext_vector_type(2))) float v2f;
typedef __attribute__((ext_vector_type(8))) float v8f;

#define DIM 128
#define NEG_SLOPE 0.01f

__device__ __forceinline__ float leaky1(float x) { return x >= 0.0f ? x : NEG_SLOPE * x; }

// Native f32 global atomic add (relaxed, agent scope -> global_atomic_add_f32)
__device__ __forceinline__ void atomAddF(float* p, float v) {
    __hip_atomic_fetch_add(p, v, __ATOMIC_RELAXED, __HIP_MEMORY_SCOPE_AGENT);
}

// ---------------------------------------------------------------- zero fill (float4)
__global__ void k_zero4(float4* __restrict__ p, long n4) {
    long i = (long)blockIdx.x * blockDim.x + threadIdx.x;
    if (i < n4) p[i] = make_float4(0.f, 0.f, 0.f, 0.f);
}

__global__ void k_zero(float* __restrict__ p, long n) {
    long i = (long)blockIdx.x * blockDim.x + threadIdx.x;
    if (i < n) p[i] = 0.0f;
}

// ---------------------------------------------------------------- degrees
__global__ void k_deg2(const int* __restrict__ src, const int* __restrict__ dst,
                       float* __restrict__ degs, float* __restrict__ degd, int E) {
    int e = blockIdx.x * blockDim.x + threadIdx.x;
    if (e < E) {
        atomAddF(&degs[src[e]], 1.0f);
        atomAddF(&degd[dst[e]], 1.0f);
    }
}

__global__ void k_wsum(const int* __restrict__ dst, const float* __restrict__ w,
                       float* __restrict__ wsum, int E) {
    int e = blockIdx.x * blockDim.x + threadIdx.x;
    if (e < E) atomAddF(&wsum[dst[e]], w[e]);
}

// ---------------------------------------------------------------- WMMA GEMM
// H[M x 128] = scale(X) @ W, W row-major [128 x 128].
// One wave per 16-row tile; 8 accumulators cover all 128 output columns.
// A layout (16x4 f32): lanes 0-15 -> K = k0,k0+1 ; lanes 16-31 -> K = k0+2,k0+3; M = lane%16.
// B layout (4x16 f32): mirrored; N = lane%16.
// C/D layout: VGPR r -> M = r + 8*(lane>=16), N = lane%16.
__global__ __launch_bounds__(128)
void k_gemm_wmma(const float* __restrict__ X, const float* __restrict__ W,
                 const float* __restrict__ degS,  // may be nullptr -> no row scaling
                 float* __restrict__ H, int numRB) {
    const int wave = threadIdx.x >> 5;
    const int lane = threadIdx.x & 31;
    const int rb = blockIdx.x * 4 + wave;
    if (rb >= numRB) return;

    const int row0 = rb * 16;
    const int m  = lane & 15;   // row-in-tile (A) / col-in-tile (B,C)
    const int kh = lane >> 4;   // K half-select
    const int row = row0 + m;

    float rs = 1.0f;
    if (degS) rs = rsqrtf(fmaxf(degS[row], 1.0f));

    v8f acc[8] = {};
    const float* xrow = X + (long)row * DIM;

    for (int k0 = 0; k0 < DIM; k0 += 4) {
        const int ka = k0 + 2 * kh;
        v2f a;
        a.x = xrow[ka]     * rs;
        a.y = xrow[ka + 1] * rs;
#pragma unroll
        for (int nt = 0; nt < 8; ++nt) {
            const float* wp = W + (long)ka * DIM + nt * 16 + m;
            v2f b;
            b.x = wp[0];
            b.y = wp[DIM];
            acc[nt] = __builtin_amdgcn_wmma_f32_16x16x4_f32(
                false, a, false, b, (short)0, acc[nt], false, false);
        }
    }

#pragma unroll
    for (int nt = 0; nt < 8; ++nt) {
        const int col = nt * 16 + m;
#pragma unroll
        for (int r = 0; r < 8; ++r) {
            H[(long)(row0 + r + 8 * kh) * DIM + col] = acc[nt][r];
        }
    }
}

// ---------------------------------------------------------------- edge scatter
// One wave per edge; lane handles cols [4*lane, 4*lane+4): one b128 gather + 4 atomics.
__global__ void k_scatter(const float* __restrict__ H, const int* __restrict__ src,
                          const int* __restrict__ dst, float* __restrict__ acc, int E) {
    long t = (long)blockIdx.x * blockDim.x + threadIdx.x;
    int e = (int)(t >> 5);
    int lane = (int)(t & 31);
    if (e >= E) return;
    int c = lane * 4;
    const float4 h = *(const float4*)(H + (long)src[e] * DIM + c);
    float* ap = acc + (long)dst[e] * DIM + c;
    atomAddF(ap + 0, h.x);
    atomAddF(ap + 1, h.y);
    atomAddF(ap + 2, h.z);
    atomAddF(ap + 3, h.w);
}

__global__ void k_scatter_w(const float* __restrict__ H, const int* __restrict__ src,
                            const int* __restrict__ dst, const float* __restrict__ w,
                            const float* __restrict__ wsum, float* __restrict__ acc, int E) {
    long t = (long)blockIdx.x * blockDim.x + threadIdx.x;
    int e = (int)(t >> 5);
    int lane = (int)(t & 31);
    if (e >= E) return;
    int d = dst[e];
    float wn = w[e] / fmaxf(wsum[d], 1e-9f);
    int c = lane * 4;
    const float4 h = *(const float4*)(H + (long)src[e] * DIM + c);
    float* ap = acc + (long)d * DIM + c;
    atomAddF(ap + 0, h.x * wn);
    atomAddF(ap + 1, h.y * wn);
    atomAddF(ap + 2, h.z * wn);
    atomAddF(ap + 3, h.w * wn);
}

// ---------------------------------------------------------------- finalize
// out = act( acc * (deg ? rsqrt(max(deg[row],1)) : 1) + bias[col] ), act 1 or 2 times.
// One thread per 4 elements (float4).
__global__ void k_finalize(const float4* __restrict__ acc, const float* __restrict__ deg,
                           const float4* __restrict__ bias, float4* __restrict__ out,
                           long nVec4, int doubleAct) {
    long i = (long)blockIdx.x * blockDim.x + threadIdx.x;
    if (i >= nVec4) return;
    long row = i >> 5;            // (i*4)/128
    int colv = (int)(i & 31);     // float4 column index within row
    float rs = 1.0f;
    if (deg) rs = rsqrtf(fmaxf(deg[row], 1.0f));
    float4 a = acc[i];
    float4 b = bias[colv];
    float4 v;
    v.x = leaky1(a.x * rs + b.x);
    v.y = leaky1(a.y * rs + b.y);
    v.z = leaky1(a.z * rs + b.z);
    v.w = leaky1(a.w * rs + b.w);
    if (doubleAct) {
        v.x = leaky1(v.x); v.y = leaky1(v.y); v.z = leaky1(v.z); v.w = leaky1(v.w);
    }
    out[i] = v;
}

// ---------------------------------------------------------------- launcher
static inline int cdiv_i(long a, long b) { return (int)((a + b - 1) / b); }

extern "C" void kernel_launch(void* const* d_in, const int* in_sizes, int n_in,
                              void* d_out, int out_size, void* d_ws, size_t ws_size,
                              hipStream_t stream) {
    const float* paper_x  = (const float*)d_in[0];
    // d_in[1] = snap_x (unused by reference)
    const float* W_cites  = (const float*)d_in[2];
    const float* b_cites  = (const float*)d_in[3];
    const float* W_isin   = (const float*)d_in[4];
    const float* b_isin   = (const float*)d_in[5];
    const float* W_snap   = (const float*)d_in[6];
    const float* b_snap   = (const float*)d_in[7];
    const float* snap_w   = (const float*)d_in[8];
    const int* cites_src  = (const int*)d_in[9];
    const int* cites_dst  = (const int*)d_in[10];
    const int* isin_src   = (const int*)d_in[11];
    const int* isin_dst   = (const int*)d_in[12];
    const int* snap_src   = (const int*)d_in[13];
    const int* snap_dst   = (const int*)d_in[14];

    const int N_PAPER = in_sizes[0] / DIM;
    const int N_SNAP  = in_sizes[1] / DIM;
    const int E_CITES = in_sizes[9];
    const int E_ISIN  = in_sizes[11];
    const int E_SNAP  = in_sizes[13];

    const long paperElems = (long)N_PAPER * DIM;
    const long snapElems  = (long)N_SNAP * DIM;

    // workspace layout (floats)
    float* ws = (float*)d_ws;
    float* deg_cs   = ws;                         // N_PAPER
    float* deg_cd   = deg_cs + N_PAPER;           // N_PAPER
    float* deg_is   = deg_cd + N_PAPER;           // N_PAPER
    float* deg_id   = deg_is + N_PAPER;           // N_SNAP
    float* wsum     = deg_id + N_SNAP;            // N_SNAP
    float* h_cites  = wsum + N_SNAP;              // paperElems
    float* h_isin   = h_cites + paperElems;       // paperElems
    float* acc_snap = h_isin + paperElems;        // snapElems
    float* snap_h   = acc_snap + snapElems;       // snapElems
    float* h2       = snap_h + snapElems;         // snapElems
    float* acc2     = h2 + snapElems;             // snapElems

    float* out_paper = (float*)d_out;             // accumulate in place
    float* out_snap  = out_paper + paperElems;

    const int T = 256;

    // 1) zero init (degree block is not 16B-aligned-count; use scalar there)
    long zdeg = 3L * N_PAPER + 2L * N_SNAP;
    hipLaunchKernelGGL(k_zero, dim3(cdiv_i(zdeg, T)), dim3(T), 0, stream, deg_cs, zdeg);
    hipLaunchKernelGGL(k_zero4, dim3(cdiv_i(paperElems / 4, T)), dim3(T), 0, stream,
                       (float4*)out_paper, paperElems / 4);
    hipLaunchKernelGGL(k_zero4, dim3(cdiv_i(snapElems / 4, T)), dim3(T), 0, stream,
                       (float4*)acc_snap, snapElems / 4);
    hipLaunchKernelGGL(k_zero4, dim3(cdiv_i(snapElems / 4, T)), dim3(T), 0, stream,
                       (float4*)acc2, snapElems / 4);

    // 2) degrees / weight sums
    hipLaunchKernelGGL(k_deg2, dim3(cdiv_i(E_CITES, T)), dim3(T), 0, stream,
                       cites_src, cites_dst, deg_cs, deg_cd, E_CITES);
    hipLaunchKernelGGL(k_deg2, dim3(cdiv_i(E_ISIN, T)), dim3(T), 0, stream,
                       isin_src, isin_dst, deg_is, deg_id, E_ISIN);
    hipLaunchKernelGGL(k_wsum, dim3(cdiv_i(E_SNAP, T)), dim3(T), 0, stream,
                       snap_dst, snap_w, wsum, E_SNAP);

    // 3) stage-1 GEMMs (row scaling fused into A-operand)
    const int rbPaper = N_PAPER / 16;
    hipLaunchKernelGGL(k_gemm_wmma, dim3(cdiv_i(rbPaper, 4)), dim3(128), 0, stream,
                       paper_x, W_cites, deg_cs, h_cites, rbPaper);
    hipLaunchKernelGGL(k_gemm_wmma, dim3(cdiv_i(rbPaper, 4)), dim3(128), 0, stream,
                       paper_x, W_isin, deg_is, h_isin, rbPaper);

    // 4) edge scatters (32 lanes per edge, float4 gather + f32 atomics)
    hipLaunchKernelGGL(k_scatter, dim3(cdiv_i((long)E_CITES * 32, T)), dim3(T), 0, stream,
                       h_cites, cites_src, cites_dst, out_paper, E_CITES);
    hipLaunchKernelGGL(k_scatter, dim3(cdiv_i((long)E_ISIN * 32, T)), dim3(T), 0, stream,
                       h_isin, isin_src, isin_dst, acc_snap, E_ISIN);

    // 5) finalize paper: rsqrt(deg_dst)*acc + b, leaky twice (etype act + final act)
    hipLaunchKernelGGL(k_finalize, dim3(cdiv_i(paperElems / 4, T)), dim3(T), 0, stream,
                       (const float4*)out_paper, deg_cd, (const float4*)b_cites,
                       (float4*)out_paper, paperElems / 4, 1);
    // snap stage-1 activation (single leaky; final leaky comes after stage 2)
    hipLaunchKernelGGL(k_finalize, dim3(cdiv_i(snapElems / 4, T)), dim3(T), 0, stream,
                       (const float4*)acc_snap, deg_id, (const float4*)b_isin,
                       (float4*)snap_h, snapElems / 4, 0);

    // 6) stage-2 weighted snap conv
    const int rbSnap = N_SNAP / 16;
    hipLaunchKernelGGL(k_gemm_wmma, dim3(cdiv_i(rbSnap, 4)), dim3(128), 0, stream,
                       snap_h, W_snap, (const float*)nullptr, h2, rbSnap);
    hipLaunchKernelGGL(k_scatter_w, dim3(cdiv_i((long)E_SNAP * 32, T)), dim3(T), 0, stream,
                       h2, snap_src, snap_dst, snap_w, wsum, acc2, E_SNAP);
    hipLaunchKernelGGL(k_finalize, dim3(cdiv_i(snapElems / 4, T)), dim3(T), 0, stream,
                       (const float4*)acc2, (const float*)nullptr, (const float4*)b_snap,
                       (float4*)out_snap, snapElems / 4, 1);
}